// HGT_41162966564911
// MI455X (gfx1250) — compile-verified
//
#include <hip/hip_runtime.h>
#include <math.h>

typedef float v2f __attribute__((ext_vector_type(2)));
typedef float v8f __attribute__((ext_vector_type(8)));

#define NA 50000
#define NP 100000
#define NE 200000
#define NHEADS 2
#define WPB 8           // waves per block in the GEMM
#define KMAX 128        // largest K we ever run (LDS B tile rows)

__device__ __forceinline__ float gelu_f(float x) {
  const float c0 = 0.7978845608028654f;  // sqrt(2/pi)
  const float c1 = 0.044715f;
  return 0.5f * x * (1.0f + tanhf(c0 * (x + c1 * x * x * x)));
}

__global__ void fill_f32(float* __restrict__ p, float v, int n) {
  int i = blockIdx.x * blockDim.x + threadIdx.x;
  if (i < n) p[i] = v;
}

__global__ void gelu_inplace(float* __restrict__ p, int n) {
  int i = blockIdx.x * blockDim.x + threadIdx.x;
  if (i < n) p[i] = gelu_f(p[i]);
}

// Batched C[z] = epilogue( A[z] @ B[z] + bias ).
// flags&2: out = sigmoid(skip)*v + (1-sigmoid)*resid[row,col], then ReLU.
// One wave computes one 16x16 tile of C with V_WMMA_F32_16X16X4_F32.
// B tile (K x 16) is staged in LDS once per block and shared by all 8 waves;
// the A fragment load is software-pipelined one K-step ahead.
__global__ void gemm_wmma_f32(const float* __restrict__ A, int lda, long long sA,
                              const float* __restrict__ B, int ldb, long long sB,
                              const float* __restrict__ bias,
                              const float* __restrict__ resid, int ldr,
                              const float* __restrict__ skipPtr,
                              float* __restrict__ C, int ldc, long long sC,
                              int M, int N, int K, int flags) {
  __shared__ float Bs[KMAX * 16];

  int nt = blockIdx.y;
  int z  = blockIdx.z;
  const float* Ab = A + (long long)z * sA;
  const float* Bb = B + (long long)z * sB;
  float*       Cb = C + (long long)z * sC;

  // ---- cooperative B-tile stage: rows 0..K-1, cols nt*16..nt*16+15 ----
  for (int i = threadIdx.x * 4; i < K * 16; i += WPB * 32 * 4) {
    int k = i >> 4, c = i & 15;
    float4 v4 = *(const float4*)(Bb + (size_t)k * ldb + nt * 16 + c);
    Bs[i + 0] = v4.x; Bs[i + 1] = v4.y; Bs[i + 2] = v4.z; Bs[i + 3] = v4.w;
  }
  __syncthreads();

  int lane = threadIdx.x & 31;
  int wid  = threadIdx.x >> 5;
  int MT   = M >> 4;
  int mt   = blockIdx.x * WPB + wid;
  if (mt >= MT) return;                 // wave-uniform: EXEC stays all-ones
  int half = lane >> 4;                 // lanes 16-31 hold K+2 / M+8 halves
  int r    = lane & 15;

  // A 16x4 f32 layout: lane<16 -> {K=0,K=1}, lane>=16 -> {K=2,K=3}, row = lane%16
  const float* arow = Ab + (size_t)(mt * 16 + r) * lda + 2 * half;
  const float* brow = Bs + 2 * half * 16 + r;   // B mirror layout out of LDS

  v8f acc = {0.f, 0.f, 0.f, 0.f, 0.f, 0.f, 0.f, 0.f};
  v2f a;
  a.x = arow[0];
  a.y = arow[1];
  for (int k = 0; k < K; k += 4) {
    int kp = (k + 4 < K) ? (k + 4) : k;         // branchless one-step prefetch
    v2f an;
    an.x = arow[kp];
    an.y = arow[kp + 1];
    v2f b;
    b.x = brow[k * 16];
    b.y = brow[k * 16 + 16];
    acc = __builtin_amdgcn_wmma_f32_16x16x4_f32(false, a, false, b, (short)0, acc,
                                                false, false);
    a = an;
  }

  int   col = nt * 16 + r;
  float bv  = bias ? bias[col] : 0.f;
  float s = 0.f, oms = 0.f;
  if (flags & 2) { s = 1.f / (1.f + expf(-skipPtr[0])); oms = 1.f - s; }
  // C/D layout: VGPR i -> row = i + 8*(lane>=16), col = lane%16
  for (int i = 0; i < 8; i++) {
    int   row = mt * 16 + i + 8 * half;
    float v   = acc[i] + bv;
    if (flags & 2) {
      v = s * v + oms * resid[(size_t)row * ldr + col];
      v = fmaxf(v, 0.f);
    }
    Cb[(size_t)row * ldc + col] = v;
  }
}

// float atomic max via monotone signed-max / unsigned-min trick
__device__ __forceinline__ void atomicMaxF(float* addr, float val) {
  if (val >= 0.f) atomicMax((int*)addr, __float_as_int(val));
  else            atomicMin((unsigned int*)addr, (unsigned int)__float_as_int(val));
}

__global__ void edge_alpha(const int* __restrict__ ei, const float* __restrict__ Q,
                           const float* __restrict__ KT, const float* __restrict__ p_rel,
                           float scale, float* __restrict__ alpha,
                           float* __restrict__ amax, int D) {
  int t = blockIdx.x * blockDim.x + threadIdx.x;
  if (t >= NE * NHEADS) return;
  int e = t / NHEADS, h = t % NHEADS;
  int src = ei[e], dst = ei[NE + e];
  const float4* qp = (const float4*)(Q + ((size_t)dst * NHEADS + h) * D);
  const float4* kp = (const float4*)(KT + ((size_t)src * NHEADS + h) * D);
  float s  = 0.f;
  int   n4 = D >> 2;
  for (int i = 0; i < n4; i++) {
    float4 q4 = qp[i], k4 = kp[i];
    s += q4.x * k4.x + q4.y * k4.y + q4.z * k4.z + q4.w * k4.w;
  }
  s *= p_rel[h] * scale;
  alpha[t] = s;
  atomicMaxF(&amax[(size_t)dst * NHEADS + h], s);
}

__global__ void edge_exp(const int* __restrict__ ei, float* __restrict__ alpha,
                         const float* __restrict__ amax, float* __restrict__ denom) {
  int t = blockIdx.x * blockDim.x + threadIdx.x;
  if (t >= NE * NHEADS) return;
  int e = t / NHEADS, h = t % NHEADS;
  int dst = ei[NE + e];
  float ex = expf(alpha[t] - amax[(size_t)dst * NHEADS + h]);
  alpha[t] = ex;  // in-place: each slot touched by exactly one thread
  atomicAdd(&denom[(size_t)dst * NHEADS + h], ex);
}

__global__ void edge_msg(const int* __restrict__ ei, const float* __restrict__ exv,
                         const float* __restrict__ denom, const float* __restrict__ VT,
                         float* __restrict__ agg, int D) {
  int t = blockIdx.x * blockDim.x + threadIdx.x;
  if (t >= NE * NHEADS) return;
  int e = t / NHEADS, h = t % NHEADS;
  int src = ei[e], dst = ei[NE + e];
  float attn = exv[t] / (denom[(size_t)dst * NHEADS + h] + 1e-16f);
  const float* vp = VT + ((size_t)src * NHEADS + h) * D;
  float*       ap = agg + ((size_t)dst * NHEADS + h) * D;
  for (int d = 0; d < D; d++) atomicAdd(&ap[d], vp[d] * attn);
}

extern "C" void kernel_launch(void* const* d_in, const int* in_sizes, int n_in,
                              void* d_out, int out_size, void* d_ws, size_t ws_size,
                              hipStream_t stream) {
  (void)in_sizes; (void)n_in; (void)out_size; (void)ws_size;

  const float* x_a  = (const float*)d_in[0];
  const float* x_p  = (const float*)d_in[1];
  const int*   ei_w = (const int*)d_in[2];
  const int*   ei_r = (const int*)d_in[3];
  const int*   ei_c = (const int*)d_in[4];

  // params flattened via JAX sorted-key tree order after the 5 positional arrays:
  // layer{1,2} -> edge{writes, cites, rev_writes}(alphabetical) x {a_rel,m_rel,p_rel}
  //            -> node{author,paper} x {a_b,a_w,k_b,k_w,q_b,q_w,skip,v_b,v_w}
  auto P       = [&](int i) { return (const float*)d_in[5 + i]; };
  auto e_a_rel = [&](int l, int et) { return P(27 * l + et * 3 + 0); };
  auto e_m_rel = [&](int l, int et) { return P(27 * l + et * 3 + 1); };
  auto e_p_rel = [&](int l, int et) { return P(27 * l + et * 3 + 2); };
  // field: 0=a_b 1=a_w 2=k_b 3=k_w 4=q_b 5=q_w 6=skip 7=v_b 8=v_w ; nt: 0=author 1=paper
  auto n_prm   = [&](int l, int nt, int f) { return P(27 * l + 9 + nt * 9 + f); };

  // ---- workspace carve (sized for layer-1 dout=128; layer 2 reuses) ----
  char* wp   = (char*)d_ws;
  auto  take = [&](size_t nf) {
    float* r = (float*)wp;
    wp += ((nf * 4 + 255) & ~(size_t)255);
    return r;
  };
  float* Qa  = take((size_t)NA * 128);
  float* Qp  = take((size_t)NP * 128);
  float* Ks  = take((size_t)NP * 128);
  float* Vs  = take((size_t)NP * 128);
  float* Kt  = take((size_t)NP * 128);
  float* Vt  = take((size_t)NP * 128);
  float* Ex  = take((size_t)NE * NHEADS);
  float* Am  = take((size_t)NP * NHEADS);
  float* Dn  = take((size_t)NP * NHEADS);
  float* AgA = take((size_t)NA * 128);
  float* AgP = take((size_t)NP * 128);
  float* Ha  = take((size_t)NA * 128);
  float* Hp  = take((size_t)NP * 128);

  auto fill = [&](float* p, float v, size_t n) {
    fill_f32<<<dim3((unsigned)((n + 255) / 256)), 256, 0, stream>>>(p, v, (int)n);
  };
  auto gelu = [&](float* p, size_t n) {
    gelu_inplace<<<dim3((unsigned)((n + 255) / 256)), 256, 0, stream>>>(p, (int)n);
  };
  auto gemm = [&](const float* A, int lda, long long sA, const float* B, int ldb,
                  long long sB, const float* bias, const float* resid, int ldr,
                  const float* skip, float* C, int ldc, long long sC, int M, int N,
                  int K, int batch, int flags) {
    dim3 g((unsigned)((M / 16 + WPB - 1) / WPB), (unsigned)(N / 16), (unsigned)batch);
    gemm_wmma_f32<<<g, WPB * 32, 0, stream>>>(A, lda, sA, B, ldb, sB, bias, resid,
                                              ldr, skip, C, ldc, sC, M, N, K, flags);
  };

  int eb = (NE * NHEADS + 255) / 256;

  auto run_layer = [&](int l, const float* ina, const float* inp, int din, float* outa,
                       float* outp, int dout, bool skiprelu) {
    int   D     = dout / NHEADS;
    float scale = 1.0f / sqrtf((float)D);

    // Q projections (dst side, needed for all edge types)
    gemm(ina, din, 0, n_prm(l, 0, 5), dout, 0, n_prm(l, 0, 4), nullptr, 0, nullptr,
         Qa, dout, 0, NA, dout, din, 1, 0);
    gemm(inp, din, 0, n_prm(l, 1, 5), dout, 0, n_prm(l, 1, 4), nullptr, 0, nullptr,
         Qp, dout, 0, NP, dout, din, 1, 0);
    fill(AgA, 0.f, (size_t)NA * dout);
    fill(AgP, 0.f, (size_t)NP * dout);

    auto run_et = [&](const int* ei, int et, int nsrc, const float* Qdst, int ndst,
                      float* agg) {
      // per-head relation transforms: kt = K_src @ a_rel[h], vt = V_src @ m_rel[h]
      gemm(Ks, dout, D, e_a_rel(l, et), D, (long long)D * D, nullptr, nullptr, 0,
           nullptr, Kt, dout, D, nsrc, D, D, NHEADS, 0);
      gemm(Vs, dout, D, e_m_rel(l, et), D, (long long)D * D, nullptr, nullptr, 0,
           nullptr, Vt, dout, D, nsrc, D, D, NHEADS, 0);
      fill(Am, -INFINITY, (size_t)ndst * NHEADS);
      fill(Dn, 0.f, (size_t)ndst * NHEADS);
      edge_alpha<<<eb, 256, 0, stream>>>(ei, Qdst, Kt, e_p_rel(l, et), scale, Ex, Am, D);
      edge_exp<<<eb, 256, 0, stream>>>(ei, Ex, Am, Dn);
      edge_msg<<<eb, 256, 0, stream>>>(ei, Ex, Dn, Vt, agg, D);
    };

    // src = author: K/V, then writes (author -> paper)
    gemm(ina, din, 0, n_prm(l, 0, 3), dout, 0, n_prm(l, 0, 2), nullptr, 0, nullptr,
         Ks, dout, 0, NA, dout, din, 1, 0);
    gemm(ina, din, 0, n_prm(l, 0, 8), dout, 0, n_prm(l, 0, 7), nullptr, 0, nullptr,
         Vs, dout, 0, NA, dout, din, 1, 0);
    run_et(ei_w, 0, NA, Qp, NP, AgP);

    // src = paper: K/V, then cites (paper -> paper) and rev_writes (paper -> author)
    gemm(inp, din, 0, n_prm(l, 1, 3), dout, 0, n_prm(l, 1, 2), nullptr, 0, nullptr,
         Ks, dout, 0, NP, dout, din, 1, 0);
    gemm(inp, din, 0, n_prm(l, 1, 8), dout, 0, n_prm(l, 1, 7), nullptr, 0, nullptr,
         Vs, dout, 0, NP, dout, din, 1, 0);
    run_et(ei_c, 1, NP, Qp, NP, AgP);
    run_et(ei_r, 2, NP, Qa, NA, AgA);

    // out = gelu(agg) @ a_w + a_b  (+ sigmoid-skip with input, + ReLU for layer 1)
    gelu(AgA, (size_t)NA * dout);
    gelu(AgP, (size_t)NP * dout);
    int flags = skiprelu ? 2 : 0;
    gemm(AgA, dout, 0, n_prm(l, 0, 1), dout, 0, n_prm(l, 0, 0), ina, din,
         n_prm(l, 0, 6), outa, dout, 0, NA, dout, dout, 1, flags);
    gemm(AgP, dout, 0, n_prm(l, 1, 1), dout, 0, n_prm(l, 1, 0), inp, din,
         n_prm(l, 1, 6), outp, dout, 0, NP, dout, dout, 1, flags);
  };

  // layer 1: 128 -> 128 (skip applies: dims match), ReLU between layers
  run_layer(0, x_a, x_p, 128, Ha, Hp, 128, true);
  // layer 2: 128 -> 64 (no skip, no relu), write straight to d_out (author then paper)
  float* oa = (float*)d_out;
  float* op = oa + (size_t)NA * 64;
  run_layer(1, Ha, Hp, 128, oa, op, 64, false);
}